// CaLCS_37838661877875
// MI455X (gfx1250) — compile-verified
//
#include <hip/hip_runtime.h>
#include <hip/hip_bf16.h>
#include <math.h>

// Problem constants from the reference
#define BB 20
#define LL 20
#define VV 100000
// padded per-batch stride in dwords: 401 mod 64 = 17 (odd) -> no LDS bank
// conflicts when 20 lanes read P[b][j][k] with b = lane.
#define PSTRIDE 401

#if defined(__has_builtin)
#if __has_builtin(__builtin_amdgcn_global_load_async_to_lds_b32)
#define USE_ASYNC_GATHER 1
#endif
#endif

typedef __attribute__((address_space(1))) int glob_i32;
typedef __attribute__((address_space(3))) int lds_i32;

__global__ __launch_bounds__(256)
void calcs_kernel(const float* __restrict__ topic_prob,
                  const int* __restrict__ hard_label,
                  float* __restrict__ out)
{
    __shared__ int   s_lbl[BB * LL];
    __shared__ float s_P[BB * PSTRIDE];

    const int tid = threadIdx.x;

    // ---- Phase 1: stage labels into LDS ----
    for (int t = tid; t < BB * LL; t += 256) {
        s_lbl[t] = hard_label[t];
    }
    __syncthreads();

    // ---- Phase 2: gather P[b][j][k] = topic_prob[b, j, clamp(lbl[b][k], 0, V-1)] ----
    // 8000 data-dependent scalar gathers, ~32 independent copies in flight per thread.
    for (int t = tid; t < BB * LL * LL; t += 256) {
        int b = t / (LL * LL);
        int r = t - b * (LL * LL);
        int j = r / LL;
        int k = r - j * LL;
        int lbl = s_lbl[b * LL + k];
        int idx = lbl < 0 ? 0 : (lbl > (VV - 1) ? (VV - 1) : lbl);
        const float* g = topic_prob + ((size_t)(b * LL + j) * (size_t)VV + (size_t)idx);
        float* l = &s_P[b * PSTRIDE + j * LL + k];
#if USE_ASYNC_GATHER
        // CDNA5 async global->LDS copy (per-lane global addr + per-lane LDS addr),
        // tracked with ASYNCcnt; no VGPR round-trip.
        __builtin_amdgcn_global_load_async_to_lds_b32(
            (glob_i32*)(void*)g, (lds_i32*)(void*)l, 0, 0);
#else
        *l = *g;
#endif
    }
#if USE_ASYNC_GATHER
    asm volatile("s_wait_asynccnt 0" ::: "memory");
#endif
    __syncthreads();

    // ---- Phase 3: DP, one lane per batch (wave 0 only; wave32) ----
    if (tid < 32) {
        const bool valid = tid < BB;
        const int  b = valid ? tid : 0;

        float mk[LL];
        int   len = 0;
#pragma unroll
        for (int k = 0; k < LL; ++k) {
            int lbl = s_lbl[b * LL + k];
            bool m = (lbl >= 0);
            mk[k] = m ? 1.0f : 0.0f;
            len  += m ? 1 : 0;
        }

        float prev[LL + 1];
#pragma unroll
        for (int i = 0; i <= LL; ++i) prev[i] = 0.0f;

        float finalv = 0.0f;   // dp[len][len]
        float colv   = 0.0f;   // current row's value at column == len

        for (int j = 0; j < LL; ++j) {
            float mj = (s_lbl[b * LL + j] >= 0) ? 1.0f : 0.0f;
            const float* prow = &s_P[b * PSTRIDE + j * LL];
            float left = 0.0f;
            float curr[LL + 1];
            curr[0] = 0.0f;
#pragma unroll
            for (int k = 0; k < LL; ++k) {
                float p    = prow[k];
                float diag = prev[k];
                float up   = prev[k + 1];
                float val  = p * (diag + 1.0f) + (1.0f - p) * fmaxf(left, up);
                val = (mj * mk[k] > 0.0f) ? val : 0.0f;
                curr[k + 1] = val;
                left = val;
                if (k + 1 == len) colv = val;   // capture dp[j+1][len]
            }
            if (j + 1 == len) finalv = colv;    // capture dp[len][len]
#pragma unroll
            for (int i = 0; i <= LL; ++i) prev[i] = curr[i];
        }

        // calcs[b] = -log(final / len); lanes >= B contribute 0
        float c = valid ? -logf(finalv / (float)len) : 0.0f;

        // wave32 reduction
#pragma unroll
        for (int off = 16; off > 0; off >>= 1) {
            c += __shfl_down(c, off, 32);
        }
        if (tid == 0) {
            out[0] = c * (1.0f / (float)BB);
        }
    }
}

extern "C" void kernel_launch(void* const* d_in, const int* in_sizes, int n_in,
                              void* d_out, int out_size, void* d_ws, size_t ws_size,
                              hipStream_t stream)
{
    const float* topic_prob = (const float*)d_in[0];
    const int*   hard_label = (const int*)d_in[1];
    float*       out        = (float*)d_out;

    hipLaunchKernelGGL(calcs_kernel, dim3(1), dim3(256), 0, stream,
                       topic_prob, hard_label, out);
}